// GHMC_Loss_57157424775631
// MI455X (gfx1250) — compile-verified
//
#include <hip/hip_runtime.h>
#include <stdint.h>

#define BINS 10
#define NCOLS 512
#define WAVES_PER_BLOCK 8
#define THREADS (WAVES_PER_BLOCK * 32)
#define PHASE1_BLOCKS 2048

// ---------------------------------------------------------------------------
// CDNA5 async global->LDS copy (gfx1250). One instruction moves 16B/lane =
// 512B/wave, tracked by ASYNCcnt; in-order completion == LDS write landed.
// Non-temporal hint: x is a 512MB single-use stream, keep it out of L2.
// ---------------------------------------------------------------------------
__device__ __forceinline__ void async_b128_to_lds(uint32_t lds_byte_off, uint64_t gaddr) {
#if defined(__gfx1250__)
  asm volatile("global_load_async_to_lds_b128 %0, %1, off th:TH_LOAD_NT"
               :: "v"(lds_byte_off), "v"(gaddr)
               : "memory");
#endif
}

// asm form (not the builtin): the "memory" clobber pins LDS reads behind the
// wait so the scheduler can't hoist ds_load_b128 above s_wait_asynccnt.
#define WAIT_ASYNC(n) asm volatile("s_wait_asynccnt %0" :: "n"(n) : "memory")

// low 32 bits of a flat pointer into LDS == workgroup-relative LDS byte offset
__device__ __forceinline__ uint32_t lds_offset_of(const void* p) {
  return (uint32_t)(uintptr_t)p;
}

// ---------------------------------------------------------------------------
// Phase 1: per-row softmax stats -> per-bin {count, ce_sum} histogram (20 f32)
// One wave (32 lanes) per row; rows double-buffered through LDS via async DMA.
// ---------------------------------------------------------------------------
__global__ __launch_bounds__(THREADS) void ghm_phase1(
    const float* __restrict__ x, const int* __restrict__ target,
    int N, float* __restrict__ ghist) {
  __shared__ float sbuf[WAVES_PER_BLOCK][2][NCOLS];   // 32 KB
  __shared__ float shist[2 * BINS];

  const int tid  = threadIdx.x;
  const int lane = tid & 31;
  const int wv   = tid >> 5;

  const int gw = blockIdx.x * WAVES_PER_BLOCK + wv;   // global wave id
  const int nw = gridDim.x * WAVES_PER_BLOCK;         // total waves

  const uint32_t lds0 = lds_offset_of(&sbuf[wv][0][0]);
  const uint32_t lds1 = lds_offset_of(&sbuf[wv][1][0]);

  // ---- prime the DMA pipeline FIRST: stage row 'gw' into buffer 0 ----
  if (gw < N) {
    uint64_t src = (uint64_t)(uintptr_t)(x + (size_t)gw * NCOLS);
#pragma unroll
    for (int j = 0; j < 4; ++j)
      async_b128_to_lds(lds0 + (uint32_t)(lane * 16 + j * 512),
                        src + (uint64_t)(lane * 16 + j * 512));
  }

  if (tid < 2 * BINS) shist[tid] = 0.0f;
  __syncthreads();

  int cur = 0;
  for (int row = gw; row < N; row += nw) {
    const int nxt = row + nw;
    const uint32_t lds_nxt = cur ? lds0 : lds1;

    if (nxt < N) {
      // issue prefetch of the next row, then wait for the current row's 4
      // async ops (in-order completion: ASYNCcnt <= 4 means cur is ready)
      uint64_t src = (uint64_t)(uintptr_t)(x + (size_t)nxt * NCOLS);
#pragma unroll
      for (int j = 0; j < 4; ++j)
        async_b128_to_lds(lds_nxt + (uint32_t)(lane * 16 + j * 512),
                          src + (uint64_t)(lane * 16 + j * 512));
      WAIT_ASYNC(4);
    } else {
      WAIT_ASYNC(0);
    }

    const float* buf = &sbuf[wv][cur][0];

    // each lane grabs its 16 columns once: cols {4*lane + 128*j .. +3}
    float vals[16];
#pragma unroll
    for (int j = 0; j < 4; ++j) {
      float4 q = *(const float4*)(buf + lane * 4 + j * 128);
      vals[4 * j + 0] = q.x;
      vals[4 * j + 1] = q.y;
      vals[4 * j + 2] = q.z;
      vals[4 * j + 3] = q.w;
    }

    // wave32 row max (butterfly over 32 lanes)
    float m = -3.402823466e+38f;
#pragma unroll
    for (int k = 0; k < 16; ++k) m = fmaxf(m, vals[k]);
#pragma unroll
    for (int off = 16; off > 0; off >>= 1) m = fmaxf(m, __shfl_xor(m, off, 32));

    // wave32 sum of exp(x - m)
    float s = 0.0f;
#pragma unroll
    for (int k = 0; k < 16; ++k) s += __expf(vals[k] - m);
#pragma unroll
    for (int off = 16; off > 0; off >>= 1) s += __shfl_xor(s, off, 32);

    const int t = __builtin_nontemporal_load(&target[row]);
    const float xt = buf[t];                 // LDS broadcast read
    const float log_pt = xt - m - __logf(s);
    const float ce = -log_pt;
    const float g = fabsf(__expf(log_pt) - 1.0f);
    int b = (int)(g * 9.9999f);              // floor(g * (BINS - 1e-4))
    b = min(max(b, 0), BINS - 1);

    if (lane == 0) {
      atomicAdd(&shist[b], 1.0f);            // ds_add_f32
      atomicAdd(&shist[BINS + b], ce);       // ds_add_f32
    }
    cur ^= 1;
  }

  __syncthreads();
  if (tid < 2 * BINS) atomicAdd(&ghist[tid], shist[tid]);  // global f32 atomic
}

// ---------------------------------------------------------------------------
// Workspace init (harness poisons ws once; we must zero it every launch)
// ---------------------------------------------------------------------------
__global__ void ghm_init(float* __restrict__ ghist) {
  if (threadIdx.x < 2 * BINS) ghist[threadIdx.x] = 0.0f;
}

// ---------------------------------------------------------------------------
// Phase 2: 20 floats -> scalar loss
// loss = (1/N) * sum_b ce_sum[b] / max(count[b] * nonempty, 1e-4)
// ---------------------------------------------------------------------------
__global__ void ghm_finalize(const float* __restrict__ ghist,
                             float* __restrict__ out, float invN) {
  if (threadIdx.x == 0) {
    float ne = 0.0f;
#pragma unroll
    for (int b = 0; b < BINS; ++b) ne += (ghist[b] > 0.0f) ? 1.0f : 0.0f;
    float loss = 0.0f;
#pragma unroll
    for (int b = 0; b < BINS; ++b) {
      float gd = fmaxf(ghist[b] * ne, 0.0001f);
      loss += ghist[BINS + b] / gd;
    }
    out[0] = loss * invN;
  }
}

// ---------------------------------------------------------------------------
extern "C" void kernel_launch(void* const* d_in, const int* in_sizes, int n_in,
                              void* d_out, int out_size, void* d_ws, size_t ws_size,
                              hipStream_t stream) {
  const float* x      = (const float*)d_in[0];
  const int*   target = (const int*)d_in[1];
  const int    N      = in_sizes[1];   // 262144 rows; in_sizes[0] = N * 512

  float* ghist = (float*)d_ws;         // 20 floats: [0..9]=counts, [10..19]=ce sums

  ghm_init<<<1, 32, 0, stream>>>(ghist);
  ghm_phase1<<<PHASE1_BLOCKS, THREADS, 0, stream>>>(x, target, N, ghist);
  ghm_finalize<<<1, 32, 0, stream>>>(ghist, (float*)d_out, 1.0f / (float)N);
}